// RGCN_13537736917577
// MI455X (gfx1250) — compile-verified
//
#include <hip/hip_runtime.h>
#include <math.h>

typedef __attribute__((ext_vector_type(2))) float v2f;
typedef __attribute__((ext_vector_type(8))) float v8f;
typedef int v4i_n __attribute__((vector_size(16)));  // matches async-LDS builtin

#define D_DIM 128
#define W_PAD 144  // LDS row stride (floats): 16-float pad -> disjoint bank sets
                   // for the two half-wave read groups (rows k and k+2)

#if __has_builtin(__builtin_amdgcn_global_load_async_to_lds_b128) && \
    __has_builtin(__builtin_amdgcn_s_wait_asynccnt)
#define HAVE_ASYNC_LDS 1
#else
#define HAVE_ASYNC_LDS 0
#endif

// ---------------------------------------------------------------------------
// x0 = relu(entity_embedding + entity_bias)
// ---------------------------------------------------------------------------
__global__ __launch_bounds__(256) void embed_relu_kernel(
    const float* __restrict__ emb, const float* __restrict__ ebias,
    float* __restrict__ X, int total) {
  int i = blockIdx.x * 256 + threadIdx.x;
  if (i >= total) return;
  X[i] = fmaxf(emb[i] + ebias[i & (D_DIM - 1)], 0.0f);
}

// ---------------------------------------------------------------------------
// deg[n] = #edges with dst == n  (computed once; shared by both layers)
// ---------------------------------------------------------------------------
__global__ __launch_bounds__(256) void degree_kernel(
    const int* __restrict__ dst, float* __restrict__ deg, int E) {
  int i = blockIdx.x * 256 + threadIdx.x;
  if (i >= E) return;
  __hip_atomic_fetch_add(&deg[dst[i]], 1.0f, __ATOMIC_RELAXED,
                         __HIP_MEMORY_SCOPE_AGENT);
}

// ---------------------------------------------------------------------------
// HB[n, b*128 + o] = sum_i X[n,i] * bases[b,i,o]
// Block = 8 waves, owns ONE base and 8 row tiles. The 64KB base weight matrix
// is staged to LDS once (async global->LDS on CDNA5, ASYNCcnt-tracked), then
// every wave's B fragments come from ds_load instead of global memory.
// Per wave: 32 K-steps x 8 v_wmma_f32_16x16x4_f32, A fragment reused 8x.
// ---------------------------------------------------------------------------
__global__ __launch_bounds__(256) void basis_gemm_wmma(
    const float* __restrict__ X, const float* __restrict__ bases,
    float* __restrict__ HB, int N, int nRowTiles, int blocksPerBase) {
  __shared__ float Wlds[D_DIM * W_PAD];  // 128 rows x 144 floats = 72 KB

  const int lane = threadIdx.x & 31;
  const int wib = threadIdx.x >> 5;
  const int b = blockIdx.x / blocksPerBase;
  const int rtBase = (blockIdx.x % blocksPerBase) * 8;
  const float* W = bases + (size_t)b * D_DIM * D_DIM;  // [128,128] row-major

  // ---- Stage W -> LDS: 4096 float4, 16 per thread, padded row stride ----
  {
#pragma unroll
    for (int i = 0; i < 16; ++i) {
      int j4 = threadIdx.x + i * 256;  // float4 index in [0,4096)
      int row = j4 >> 5;               // 32 float4 per 128-float row
      int c4 = j4 & 31;
      const float* gsrc = W + (size_t)row * D_DIM + c4 * 4;
      float* ldst = &Wlds[row * W_PAD + c4 * 4];
#if HAVE_ASYNC_LDS
      __builtin_amdgcn_global_load_async_to_lds_b128(
          (__attribute__((address_space(1))) v4i_n*)gsrc,
          (__attribute__((address_space(3))) v4i_n*)ldst, 0, 0);
#else
      *(float4*)ldst = *(const float4*)gsrc;
#endif
    }
#if HAVE_ASYNC_LDS
    __builtin_amdgcn_s_wait_asynccnt(0);
#endif
  }
  __syncthreads();

  const int rt = rtBase + wib;
  if (rt >= nRowTiles) return;  // wave-uniform: EXEC stays all-ones for WMMA

  const int mlo = lane & 15;
  const int hi = lane >> 4;  // 0 or 1
  int mA = rt * 16 + mlo;
  if (mA >= N) mA = N - 1;   // clamp reads only (writes are guarded)
  const int khalf = hi * 2;  // lanes 16-31 hold K+2 / K+3

  v8f zero = {};
  v8f acc[8];
#pragma unroll
  for (int t = 0; t < 8; ++t) acc[t] = zero;

  const float* xrow = X + (size_t)mA * D_DIM + khalf;
  for (int k = 0; k < D_DIM; k += 4) {
    // A fragment (16x4 f32): lane holds A[m][k+khalf], A[m][k+khalf+1]
    v2f a;
    a.x = xrow[k];
    a.y = xrow[k + 1];
    const float* wrow = &Wlds[(k + khalf) * W_PAD + mlo];
#pragma unroll
    for (int t = 0; t < 8; ++t) {
      // B fragment (4x16 f32) from LDS: W[k+khalf][col], W[k+khalf+1][col]
      v2f bf;
      bf.x = wrow[t * 16];
      bf.y = wrow[t * 16 + W_PAD];
      acc[t] = __builtin_amdgcn_wmma_f32_16x16x4_f32(
          false, a, false, bf, (short)0, acc[t], false, false);
    }
  }

  // C/D layout: VGPR r, lanes 0-15 -> M=r, lanes 16-31 -> M=r+8; N=lane%16
  const int rowBase = rt * 16 + 8 * hi;
#pragma unroll
  for (int t = 0; t < 8; ++t) {
#pragma unroll
    for (int r = 0; r < 8; ++r) {
      int n = rowBase + r;
      if (n < N)
        HB[(size_t)n * (4 * D_DIM) + b * D_DIM + t * 16 + mlo] = acc[t][r];
    }
  }
}

// ---------------------------------------------------------------------------
// Per edge e: msg = sum_b comp[etype[e],b] * HB[src[e], b, :]
//             agg[dst[e], :] += msg   (f32 atomics, L2-resident target)
// One wave per edge; lane handles 4 contiguous outputs (float4 loads).
// ---------------------------------------------------------------------------
__global__ __launch_bounds__(256) void edge_msg_scatter(
    const float* __restrict__ HB, const int* __restrict__ src,
    const int* __restrict__ dst, const int* __restrict__ etype,
    const float* __restrict__ comp, float* __restrict__ agg, int E) {
  const int lane = threadIdx.x & 31;
  const int e = (blockIdx.x * 256 + threadIdx.x) >> 5;
  if (e >= E) return;
  const int s = src[e], d = dst[e], r = etype[e];
  const float c0 = comp[r * 4 + 0], c1 = comp[r * 4 + 1];
  const float c2 = comp[r * 4 + 2], c3 = comp[r * 4 + 3];
  const float4* hp = (const float4*)(HB + (size_t)s * (4 * D_DIM));
  float4 h0 = hp[lane];       // base 0
  float4 h1 = hp[32 + lane];  // base 1 (128 floats = 32 float4)
  float4 h2 = hp[64 + lane];  // base 2
  float4 h3 = hp[96 + lane];  // base 3
  float4 m;
  m.x = c0 * h0.x + c1 * h1.x + c2 * h2.x + c3 * h3.x;
  m.y = c0 * h0.y + c1 * h1.y + c2 * h2.y + c3 * h3.y;
  m.z = c0 * h0.z + c1 * h1.z + c2 * h2.z + c3 * h3.z;
  m.w = c0 * h0.w + c1 * h1.w + c2 * h2.w + c3 * h3.w;
  float* ap = agg + (size_t)d * D_DIM + lane * 4;
  __hip_atomic_fetch_add(ap + 0, m.x, __ATOMIC_RELAXED, __HIP_MEMORY_SCOPE_AGENT);
  __hip_atomic_fetch_add(ap + 1, m.y, __ATOMIC_RELAXED, __HIP_MEMORY_SCOPE_AGENT);
  __hip_atomic_fetch_add(ap + 2, m.z, __ATOMIC_RELAXED, __HIP_MEMORY_SCOPE_AGENT);
  __hip_atomic_fetch_add(ap + 3, m.w, __ATOMIC_RELAXED, __HIP_MEMORY_SCOPE_AGENT);
}

// ---------------------------------------------------------------------------
// X[n,o] = act(agg[n,o] / max(deg[n],1) + bias[o])
// ---------------------------------------------------------------------------
__global__ __launch_bounds__(256) void norm_bias_act(
    const float* __restrict__ agg, const float* __restrict__ deg,
    const float* __restrict__ bias, float* __restrict__ X, int total,
    int doRelu) {
  int i = blockIdx.x * 256 + threadIdx.x;
  if (i >= total) return;
  int n = i >> 7, o = i & (D_DIM - 1);
  float v = agg[i] / fmaxf(deg[n], 1.0f) + bias[o];
  X[i] = doRelu ? fmaxf(v, 0.0f) : v;
}

__device__ __forceinline__ float softplus_stable(float z) {
  // log(1+exp(z)), numerically stable
  return fmaxf(z, 0.0f) + log1pf(expf(-fabsf(z)));
}

// ---------------------------------------------------------------------------
// DistMult decode + loss/auc partial sums. One wave per edge; shuffle-reduce
// over 128 dims (4 per lane), block-reduce partials, one atomicAdd per block.
// ---------------------------------------------------------------------------
__global__ __launch_bounds__(256) void distmult_kernel(
    const float* __restrict__ X, const int* __restrict__ h,
    const int* __restrict__ t, const int* __restrict__ et,
    const int* __restrict__ nt, const float* __restrict__ rel,
    float* __restrict__ pred, float* __restrict__ acc, int E) {
  __shared__ float s_lp[8], s_ln[8], s_auc[8];
  const int lane = threadIdx.x & 31;
  const int wib = threadIdx.x >> 5;
  const int e = (blockIdx.x * 256 + threadIdx.x) >> 5;
  float lp = 0.0f, ln = 0.0f, au = 0.0f;
  if (e < E) {
    const float4* xh = (const float4*)(X + (size_t)h[e] * D_DIM);
    const float4* xt = (const float4*)(X + (size_t)t[e] * D_DIM);
    const float4* xn = (const float4*)(X + (size_t)nt[e] * D_DIM);
    const float4* rr = (const float4*)(rel + (size_t)et[e] * D_DIM);
    float4 a = xh[lane], b = rr[lane], c = xt[lane], dn = xn[lane];
    float4 hr = {a.x * b.x, a.y * b.y, a.z * b.z, a.w * b.w};
    float p = hr.x * c.x + hr.y * c.y + hr.z * c.z + hr.w * c.w;
    float q = hr.x * dn.x + hr.y * dn.y + hr.z * dn.z + hr.w * dn.w;
#pragma unroll
    for (int o = 16; o > 0; o >>= 1) {
      p += __shfl_xor(p, o, 32);
      q += __shfl_xor(q, o, 32);
    }
    if (lane == 0) {
      pred[e] = p;
      lp = softplus_stable(-p);  // -log_sigmoid(p)
      ln = softplus_stable(q);   // -log_sigmoid(-q)
      au = (p > q) ? 1.0f : 0.0f;
    }
  }
  if (lane == 0) { s_lp[wib] = lp; s_ln[wib] = ln; s_auc[wib] = au; }
  __syncthreads();
  if (threadIdx.x == 0) {
    float slp = 0.0f, sln = 0.0f, sau = 0.0f;
#pragma unroll
    for (int i = 0; i < 8; ++i) { slp += s_lp[i]; sln += s_ln[i]; sau += s_auc[i]; }
    __hip_atomic_fetch_add(&acc[0], slp, __ATOMIC_RELAXED, __HIP_MEMORY_SCOPE_AGENT);
    __hip_atomic_fetch_add(&acc[1], sln, __ATOMIC_RELAXED, __HIP_MEMORY_SCOPE_AGENT);
    __hip_atomic_fetch_add(&acc[2], sau, __ATOMIC_RELAXED, __HIP_MEMORY_SCOPE_AGENT);
  }
}

__global__ void finalize_kernel(const float* __restrict__ acc,
                                float* __restrict__ out, int E) {
  if (blockIdx.x == 0 && threadIdx.x == 0) {
    float invE = 1.0f / (float)E;
    out[E] = 0.5f * (acc[0] + acc[1]) * invE;  // loss
    out[E + 1] = acc[2] * invE;                // roc_auc
  }
}

// ---------------------------------------------------------------------------
extern "C" void kernel_launch(void* const* d_in, const int* in_sizes, int n_in,
                              void* d_out, int out_size, void* d_ws,
                              size_t ws_size, hipStream_t stream) {
  const float* emb    = (const float*)d_in[0];
  const float* ebias  = (const float*)d_in[1];
  const float* bases1 = (const float*)d_in[2];
  const float* comp1  = (const float*)d_in[3];
  const float* bias1  = (const float*)d_in[4];
  const float* bases2 = (const float*)d_in[5];
  const float* comp2  = (const float*)d_in[6];
  const float* bias2  = (const float*)d_in[7];
  const float* rel    = (const float*)d_in[8];
  const int* eidx     = (const int*)d_in[9];
  const int* etype    = (const int*)d_in[10];
  const int* ntail    = (const int*)d_in[11];

  const int N = in_sizes[0] / D_DIM;
  const int E = in_sizes[10];
  const int* src = eidx;      // edge_index[0]
  const int* dst = eidx + E;  // edge_index[1]

  // Workspace layout (256B-aligned slices)
  size_t off = 0;
  auto carve = [&](size_t bytes) {
    void* p = (char*)d_ws + off;
    off += (bytes + 255) & ~(size_t)255;
    return p;
  };
  float* XA  = (float*)carve((size_t)N * D_DIM * 4);      // features (in-place)
  float* HB  = (float*)carve((size_t)N * 4 * D_DIM * 4);  // basis projections
  float* AGG = (float*)carve((size_t)N * D_DIM * 4);      // scatter target
  float* DEG = (float*)carve((size_t)N * 4);              // degrees
  float* ACC = (float*)carve(64);                         // loss/auc sums
  if (off > ws_size) return;  // insufficient scratch

  const int totalND = N * D_DIM;
  const int nRowTiles = (N + 15) / 16;
  const int blocksPerBase = (nRowTiles + 7) / 8;
  const int gemmBlks = 4 * blocksPerBase;  // 4 bases
  const int edgeBlks = (E + 7) / 8;        // 1 wave/edge

  (void)hipMemsetAsync(DEG, 0, (size_t)N * 4, stream);
  (void)hipMemsetAsync(ACC, 0, 64, stream);
  (void)hipMemsetAsync(AGG, 0, (size_t)N * D_DIM * 4, stream);

  embed_relu_kernel<<<(totalND + 255) / 256, 256, 0, stream>>>(emb, ebias, XA,
                                                               totalND);
  degree_kernel<<<(E + 255) / 256, 256, 0, stream>>>(dst, DEG, E);

  // Layer 1
  basis_gemm_wmma<<<gemmBlks, 256, 0, stream>>>(XA, bases1, HB, N, nRowTiles,
                                                blocksPerBase);
  edge_msg_scatter<<<edgeBlks, 256, 0, stream>>>(HB, src, dst, etype, comp1,
                                                 AGG, E);
  norm_bias_act<<<(totalND + 255) / 256, 256, 0, stream>>>(AGG, DEG, bias1, XA,
                                                           totalND, 1);

  // Layer 2
  (void)hipMemsetAsync(AGG, 0, (size_t)N * D_DIM * 4, stream);
  basis_gemm_wmma<<<gemmBlks, 256, 0, stream>>>(XA, bases2, HB, N, nRowTiles,
                                                blocksPerBase);
  edge_msg_scatter<<<edgeBlks, 256, 0, stream>>>(HB, src, dst, etype, comp2,
                                                 AGG, E);
  norm_bias_act<<<(totalND + 255) / 256, 256, 0, stream>>>(AGG, DEG, bias2, XA,
                                                           totalND, 0);

  // Decode + loss/auc
  float* pred = (float*)d_out;
  distmult_kernel<<<edgeBlks, 256, 0, stream>>>(XA, src, dst, etype, ntail,
                                                rel, pred, ACC, E);
  finalize_kernel<<<1, 64, 0, stream>>>(ACC, pred, E);
}